// ConstraintLoss_39135742001660
// MI455X (gfx1250) — compile-verified
//
#include <hip/hip_runtime.h>
#include <math.h>

typedef __attribute__((ext_vector_type(2))) float v2f;
typedef __attribute__((ext_vector_type(8))) float v8f;

// D = A(16x4 f32) * B(4x16 f32) + C(16x16 f32), full-precision fp32 WMMA.
// args: (neg_a, A, neg_b, B, c_mod, C, reuse_a, reuse_b)
#define WMMA_F32_4(A, B, C) \
  __builtin_amdgcn_wmma_f32_16x16x4_f32(false, (A), false, (B), (short)0, (C), false, false)

// ---------------------------------------------------------------------------
// Wave32 all-lane sum via fp32 WMMA.
//   Step 1: A vgpr0 = x, vgpr1 = 0; B = ones  -> D1[m][n] = x[m] + x[m+16]
//           (each lane value lands in exactly one row; rows constant across n)
//   Step 2: A = ones; B = D1 vgpr pairs, C chained -> every element = sum of
//           all 16 row-sums = total of all 32 lane values (exact fp32 adds).
// EXEC must be all-ones on entry (callers keep the wave convergent).
// ---------------------------------------------------------------------------
__device__ __forceinline__ float wave32_sum_wmma(float x) {
  v2f a;    a.x = x;    a.y = 0.0f;
  v2f ones; ones.x = 1.0f; ones.y = 1.0f;
  v8f zero = {};
  v8f d1 = WMMA_F32_4(a, ones, zero);

  v8f acc = {};
  v2f b;
  b.x = d1[0]; b.y = d1[1];
  acc = WMMA_F32_4(ones, b, acc);
  b.x = d1[2]; b.y = d1[3];
  acc = WMMA_F32_4(ones, b, acc);
  b.x = d1[4]; b.y = d1[5];
  acc = WMMA_F32_4(ones, b, acc);
  b.x = d1[6]; b.y = d1[7];
  acc = WMMA_F32_4(ones, b, acc);
  return acc[0];
}

// ---------------------------------------------------------------------------
// Kernel 1: probs = sigmoid(pred); ax = 0; out = 0
// ---------------------------------------------------------------------------
__global__ void cl_init_kernel(const float* __restrict__ pred,
                               float* __restrict__ probs,
                               float* __restrict__ ax,
                               float* __restrict__ out,
                               int n_vars, int n_constrs) {
  int tid    = blockIdx.x * blockDim.x + threadIdx.x;
  int stride = gridDim.x * blockDim.x;
  int n = (n_vars > n_constrs) ? n_vars : n_constrs;
  for (int i = tid; i < n; i += stride) {
    if (i < n_vars) {
      float p = pred[i];
      probs[i] = 1.0f / (1.0f + expf(-p));
    }
    if (i < n_constrs) {
      ax[i] = 0.0f;
    }
  }
  if (tid == 0) out[0] = 0.0f;
}

// ---------------------------------------------------------------------------
// Kernel 2: ax[constr_idx[j]] += coeff[j] * probs[var_idx[j]]
// 240 MB of streaming reads (b128 loads) -> bandwidth bound (~10 us floor).
// probs (4MB) and ax (2MB) are L2-resident; fp atomics at agent scope lower
// to hardware global_atomic_add_f32 (non-returning).
// ---------------------------------------------------------------------------
__global__ void cl_scatter_kernel(const int*   __restrict__ cidx,
                                  const int*   __restrict__ vidx,
                                  const float* __restrict__ coeff,
                                  const float* __restrict__ probs,
                                  float*       __restrict__ ax,
                                  int nnz) {
  int tid    = blockIdx.x * blockDim.x + threadIdx.x;
  int stride = gridDim.x * blockDim.x;
  int nvec   = nnz >> 2;

  const int4*   c4 = (const int4*)cidx;
  const int4*   v4 = (const int4*)vidx;
  const float4* w4 = (const float4*)coeff;

  for (int i = tid; i < nvec; i += stride) {
    int4   ci = c4[i];
    int4   vi = v4[i];
    float4 w  = w4[i];
    float p0 = probs[vi.x];
    float p1 = probs[vi.y];
    float p2 = probs[vi.z];
    float p3 = probs[vi.w];
    __hip_atomic_fetch_add(&ax[ci.x], w.x * p0, __ATOMIC_RELAXED, __HIP_MEMORY_SCOPE_AGENT);
    __hip_atomic_fetch_add(&ax[ci.y], w.y * p1, __ATOMIC_RELAXED, __HIP_MEMORY_SCOPE_AGENT);
    __hip_atomic_fetch_add(&ax[ci.z], w.z * p2, __ATOMIC_RELAXED, __HIP_MEMORY_SCOPE_AGENT);
    __hip_atomic_fetch_add(&ax[ci.w], w.w * p3, __ATOMIC_RELAXED, __HIP_MEMORY_SCOPE_AGENT);
  }
  // tail (nnz not a multiple of 4)
  for (int i = (nvec << 2) + tid; i < nnz; i += stride) {
    float v = coeff[i] * probs[vidx[i]];
    __hip_atomic_fetch_add(&ax[cidx[i]], v, __ATOMIC_RELAXED, __HIP_MEMORY_SCOPE_AGENT);
  }
}

// ---------------------------------------------------------------------------
// Kernel 3: per-constraint violation, WMMA wave reduction, atomic into out.
// All lanes stay convergent until after the WMMA reduction (EXEC all ones).
// ---------------------------------------------------------------------------
__global__ void cl_violate_kernel(const float* __restrict__ ax,
                                  const float* __restrict__ rhs,
                                  const int*   __restrict__ sense,
                                  float*       __restrict__ out,
                                  int n_constrs, float inv_n) {
  int tid    = blockIdx.x * blockDim.x + threadIdx.x;
  int stride = gridDim.x * blockDim.x;

  float acc = 0.0f;
  for (int i = tid; i < n_constrs; i += stride) {
    float a = ax[i];
    float r = rhs[i];
    int   s = sense[i];
    float d = a - r;
    float v;
    if (s == 1)      v = fmaxf(d, 0.0f);   // <=
    else if (s == 2) v = fmaxf(-d, 0.0f);  // >=
    else if (s == 3) v = fabsf(d);         // ==
    else             v = 0.0f;
    acc += v;
  }

  // whole wave reconverged here: full-EXEC fp32 WMMA tree reduction
  float wave_total = wave32_sum_wmma(acc);

  if ((threadIdx.x & 31u) == 0u) {
    __hip_atomic_fetch_add(out, wave_total * inv_n,
                           __ATOMIC_RELAXED, __HIP_MEMORY_SCOPE_AGENT);
  }
}

// ---------------------------------------------------------------------------
// Host entry
// in order: pred, constr_idx, var_idx, coeff, constr_rhs, constr_sense,
//           n_vars(scalar), n_constrs(scalar)
// ---------------------------------------------------------------------------
extern "C" void kernel_launch(void* const* d_in, const int* in_sizes, int n_in,
                              void* d_out, int out_size, void* d_ws, size_t ws_size,
                              hipStream_t stream) {
  const float* pred  = (const float*)d_in[0];
  const int*   cidx  = (const int*)  d_in[1];
  const int*   vidx  = (const int*)  d_in[2];
  const float* coeff = (const float*)d_in[3];
  const float* rhs   = (const float*)d_in[4];
  const int*   sense = (const int*)  d_in[5];

  const int n_vars    = in_sizes[0];  // pred length
  const int nnz       = in_sizes[1];  // constr_idx length
  const int n_constrs = in_sizes[4];  // constr_rhs length

  float* probs = (float*)d_ws;          // n_vars floats   (4 MB)
  float* ax    = probs + n_vars;        // n_constrs floats (2 MB)
  float* out   = (float*)d_out;

  cl_init_kernel<<<1024, 256, 0, stream>>>(pred, probs, ax, out, n_vars, n_constrs);
  cl_scatter_kernel<<<4096, 256, 0, stream>>>(cidx, vidx, coeff, probs, ax, nnz);
  cl_violate_kernel<<<1024, 256, 0, stream>>>(ax, rhs, sense, out, n_constrs,
                                              1.0f / (float)n_constrs);
}